// InvResBlock_80401787781413
// MI455X (gfx1250) — compile-verified
//
#include <hip/hip_runtime.h>
#include <math.h>

// Problem constants (match reference)
#define Bsz   8
#define Nn    10000
#define Cc    64
#define DIMH  256
#define Ee    160000
#define BN    (Bsz * Nn)          // 80000 rows
#define BNC   (Bsz * Nn * Cc)     // 5,120,000

// padded LDS pitches (avoid 16-way bank conflicts on fragment reads)
#define ALD   66                  // pitch for 16x64 A tiles
#define HLD   258                 // pitch for 16x256 h tiles

typedef float v2f __attribute__((ext_vector_type(2)));
typedef float v8f __attribute__((ext_vector_type(8)));

// ---------------------------------------------------------------------------
// CDNA5 fp32 WMMA: D(16x16) = A(16x4) * B(4x16) + C   (v_wmma_f32_16x16x4_f32)
// ---------------------------------------------------------------------------
__device__ __forceinline__ v8f wmma_f32(v2f a, v2f b, v8f c) {
    // (neg_a, A, neg_b, B, c_mod, C, reuse_a, reuse_b)
    return __builtin_amdgcn_wmma_f32_16x16x4_f32(false, a, false, b,
                                                 (short)0, c, false, false);
}

// A fragment (16x4 tile at k-base kb) from an LDS tile with pitch lda.
// ISA layout: lane l holds row (l&15); lanes<16 -> K = kb,kb+1 ; lanes>=16 -> kb+2,kb+3
__device__ __forceinline__ v2f load_a_s(const float* As, int lda, int kb, int lane) {
    int m  = lane & 15;
    int k0 = kb + ((lane >> 4) << 1);
    const float* p = As + m * lda + k0;
    v2f a; a.x = p[0]; a.y = p[1];
    return a;
}

// B fragment from K-pair-packed weights: Bp[k2 * ldb + j] = (B[2k2][j], B[2k2+1][j]).
// One coalesced global_load_b64 per fragment.
__device__ __forceinline__ v2f load_b_p(const float2* __restrict__ Bp, int col0,
                                        int ldb, int kb, int lane) {
    int n    = lane & 15;
    int half = lane >> 4;
    float2 v = Bp[(size_t)((kb >> 1) + half) * ldb + col0 + n];
    v2f b; b.x = v.x; b.y = v.y;
    return b;
}

__device__ __forceinline__ float swish(float h, float sb) {
    return h * (1.0f / (1.0f + __expf(-h * sb))) * (1.0f / 1.1f);
}

// ---------------------------------------------------------------------------
// Graph kernels
// ---------------------------------------------------------------------------
__global__ void k_deg(const float* __restrict__ ew, const int* __restrict__ dst,
                      float* __restrict__ deg) {
    int e = blockIdx.x * blockDim.x + threadIdx.x;
    if (e < Ee) atomicAdd(&deg[dst[e]], ew[e]);
}

__global__ void k_lapw(const float* __restrict__ ew, const int* __restrict__ src,
                       const int* __restrict__ dst, const float* __restrict__ deg,
                       float* __restrict__ lapw) {
    int e = blockIdx.x * blockDim.x + threadIdx.x;
    if (e >= Ee) return;
    float ds = deg[src[e]], dd = deg[dst[e]];
    float is = ds > 0.0f ? rsqrtf(ds) : 0.0f;
    float id = dd > 0.0f ? rsqrtf(dd) : 0.0f;
    lapw[e] = -(is * ew[e] * id);
}

// Fold Chebyshev weights and K-pair-pack all three:
//   W0p = Wc[0]-Wc[2], W1p = Wc[1], W2p = 2*Wc[2]   (each packed [32][256] float2)
__global__ void k_prep_cheb(const float* __restrict__ Wc, float2* __restrict__ W0p,
                            float2* __restrict__ W1p, float2* __restrict__ W2p) {
    int i = blockIdx.x * blockDim.x + threadIdx.x;
    if (i >= (Cc / 2) * DIMH) return;
    int k2 = i / DIMH, j = i % DIMH;
    const float* w0 = Wc;
    const float* w1 = Wc + Cc * DIMH;
    const float* w2 = Wc + 2 * Cc * DIMH;
    size_t r0 = (size_t)(2 * k2) * DIMH + j;
    size_t r1 = r0 + DIMH;
    W0p[i] = make_float2(w0[r0] - w2[r0], w0[r1] - w2[r1]);
    W1p[i] = make_float2(w1[r0],          w1[r1]);
    W2p[i] = make_float2(2.0f * w2[r0],   2.0f * w2[r1]);
}

// Generic K-pair pack: Bp[k2*Nout + j] = (B[2k2][j], B[2k2+1][j])
__global__ void k_pack(const float* __restrict__ B, float2* __restrict__ Bp,
                       int K2, int Nout) {
    int i = blockIdx.x * blockDim.x + threadIdx.x;
    if (i >= K2 * Nout) return;
    int k2 = i / Nout, j = i % Nout;
    size_t r0 = (size_t)(2 * k2) * Nout + j;
    Bp[i] = make_float2(B[r0], B[r0 + Nout]);
}

// SpMM scatter: out[b, dst, :] += in[b, src, :] * lapw[e]
__global__ void k_prop(const float* __restrict__ in, float* __restrict__ out,
                       const int* __restrict__ src, const int* __restrict__ dst,
                       const float* __restrict__ lapw) {
    long long tid = (long long)blockIdx.x * blockDim.x + threadIdx.x;
    int e = (int)(tid >> 6);
    int c = (int)(tid & 63);
    if (e >= Ee) return;
    float w = lapw[e];
    const float* ip = in + (size_t)src[e] * Cc + c;
    float* op = out + (size_t)dst[e] * Cc + c;
#pragma unroll
    for (int b = 0; b < Bsz; ++b) {
        atomicAdd(op + (size_t)b * Nn * Cc, ip[(size_t)b * Nn * Cc] * w);
    }
}

// ---------------------------------------------------------------------------
// Fused MLP: one block (8 wave32) owns 16 rows.
// A tiles (Tx0/Tx1/P2) staged once in LDS; h1/h2 staged in LDS; all weights
// read as K-pair-packed float2 (one b64 per WMMA B fragment).
//   h1 = swish(Tx0@W0p + Tx1@W1p + P2@W2p + b_cheb)
//   h2 = swish(h1@W1 + b1)
//   Fx = h2@W2 + b2 ;  out0 = Fx + x ; out1 = Fx
// ---------------------------------------------------------------------------
__global__ __launch_bounds__(256) void fused_mlp(
    const float* __restrict__ Tx0, const float* __restrict__ Tx1,
    const float* __restrict__ P2,
    const float2* __restrict__ W0p, const float2* __restrict__ W1p,
    const float2* __restrict__ W2p, const float* __restrict__ bcheb,
    const float2* __restrict__ W1,  const float* __restrict__ b1,
    const float2* __restrict__ W2,  const float* __restrict__ b2,
    const float* __restrict__ beta,
    float* __restrict__ out_sum, float* __restrict__ out_fx)
{
    __shared__ float sA0[16 * ALD];   // x tile (also residual source)
    __shared__ float sA1[16 * ALD];   // Tx1 tile
    __shared__ float sA2[16 * ALD];   // P2 tile
    __shared__ float h1[16 * HLD];
    __shared__ float h2[16 * HLD];

    const int lane = threadIdx.x & 31;
    const int wave = threadIdx.x >> 5;
    const int row0 = blockIdx.x * 16;
    const float sb = __logf(1.0f + __expf(beta[0]));  // softplus(beta), uniform

    // hint the weight streams toward cache (global_prefetch_b8)
    __builtin_prefetch(W0p + lane, 0, 1);
    __builtin_prefetch(W1  + lane, 0, 1);
    __builtin_prefetch(W2  + lane, 0, 1);

    // ---- Stage 0: cooperative LDS staging of the three 16x64 A tiles ------
#pragma unroll
    for (int i = 0; i < 2; ++i) {
        int f = (threadIdx.x + i * 256) * 2;      // even float index, 0..1022
        int r = f >> 6, c = f & 63;
        size_t g = (size_t)(row0 + r) * Cc + c;
        *(float2*)&sA0[r * ALD + c] = *(const float2*)&Tx0[g];
        *(float2*)&sA1[r * ALD + c] = *(const float2*)&Tx1[g];
        *(float2*)&sA2[r * ALD + c] = *(const float2*)&P2[g];
    }
    __syncthreads();

    // ---- Stage 1: Chebyshev GEMM (K = 3x64) + swish -> LDS h1 -------------
#pragma unroll
    for (int t = 0; t < 2; ++t) {
        const int col0 = (wave * 2 + t) * 16;
        v8f a0 = {}, a1 = {}, a2 = {};   // independent accs: breaks RAW chain
#pragma unroll
        for (int kb = 0; kb < Cc; kb += 4) {
            a0 = wmma_f32(load_a_s(sA0, ALD, kb, lane),
                          load_b_p(W0p, col0, DIMH, kb, lane), a0);
            a1 = wmma_f32(load_a_s(sA1, ALD, kb, lane),
                          load_b_p(W1p, col0, DIMH, kb, lane), a1);
            a2 = wmma_f32(load_a_s(sA2, ALD, kb, lane),
                          load_b_p(W2p, col0, DIMH, kb, lane), a2);
        }
        const int half = lane >> 4, n = lane & 15;
#pragma unroll
        for (int r = 0; r < 8; ++r) {
            int m = r + 8 * half;
            float h = a0[r] + a1[r] + a2[r] + bcheb[col0 + n];
            h1[m * HLD + col0 + n] = swish(h, sb);
        }
    }
    __syncthreads();

    // ---- Stage 2: h2 = swish(h1 @ W1 + b1), K = 256 -----------------------
#pragma unroll
    for (int t = 0; t < 2; ++t) {
        const int col0 = (wave * 2 + t) * 16;
        v8f acc = {};
#pragma unroll 8
        for (int kb = 0; kb < DIMH; kb += 4) {
            acc = wmma_f32(load_a_s(h1, HLD, kb, lane),
                           load_b_p(W1, col0, DIMH, kb, lane), acc);
        }
        const int half = lane >> 4, n = lane & 15;
#pragma unroll
        for (int r = 0; r < 8; ++r) {
            int m = r + 8 * half;
            float h = acc[r] + b1[col0 + n];
            h2[m * HLD + col0 + n] = swish(h, sb);
        }
    }
    __syncthreads();

    // ---- Stage 3: Fx = h2 @ W2 + b2 (N=64, waves 0-3), residual + store ---
    if (wave < 4) {
        const int col0 = wave * 16;
        v8f acc = {};
#pragma unroll 8
        for (int kb = 0; kb < DIMH; kb += 4) {
            acc = wmma_f32(load_a_s(h2, HLD, kb, lane),
                           load_b_p(W2, col0, Cc, kb, lane), acc);
        }
        const int half = lane >> 4, n = lane & 15;
#pragma unroll
        for (int r = 0; r < 8; ++r) {
            int m = r + 8 * half;
            size_t idx = (size_t)(row0 + m) * Cc + col0 + n;
            float fx = acc[r] + b2[col0 + n];
            out_fx[idx]  = fx;
            out_sum[idx] = fx + sA0[m * ALD + col0 + n];   // residual from LDS
        }
    }
}

// ---------------------------------------------------------------------------
extern "C" void kernel_launch(void* const* d_in, const int* in_sizes, int n_in,
                              void* d_out, int out_size, void* d_ws, size_t ws_size,
                              hipStream_t stream) {
    const float* x      = (const float*)d_in[0];   // [B, N*C]
    const float* ew     = (const float*)d_in[1];   // [E]
    const float* W_cheb = (const float*)d_in[2];   // [3, 64, 256]
    const float* b_cheb = (const float*)d_in[3];   // [256]
    const float* W1     = (const float*)d_in[4];   // [256, 256]
    const float* b1     = (const float*)d_in[5];   // [256]
    const float* W2     = (const float*)d_in[6];   // [256, 64]
    const float* b2     = (const float*)d_in[7];   // [64]
    const float* beta   = (const float*)d_in[8];   // [1]
    const int*   ei     = (const int*)d_in[9];     // [2, E]
    const int* src = ei;
    const int* dst = ei + Ee;

    float* ws = (float*)d_ws;
    // Layout (floats): [Tx1: BNC][P2: BNC][deg: N]  <- zeroed together
    //                  [lapw: E][packed weights...]
    float*  Tx1   = ws;
    float*  P2    = Tx1 + BNC;
    float*  deg   = P2 + BNC;
    float*  lapw  = deg + Nn;
    float*  pw    = lapw + Ee;                     // even offset -> float2 aligned
    float2* W0p_p = (float2*)pw;                   // 32*256 float2
    float2* W1p_p = W0p_p + (Cc / 2) * DIMH;
    float2* W2p_p = W1p_p + (Cc / 2) * DIMH;
    float2* W1_p  = W2p_p + (Cc / 2) * DIMH;       // 128*256 float2
    float2* W2_p  = W1_p + (DIMH / 2) * DIMH;      // 128*64  float2

    float* out_sum = (float*)d_out;                // Fx + x
    float* out_fx  = out_sum + BNC;                // Fx

    // zero accumulators (Tx1, P2, deg are contiguous)
    hipMemsetAsync(Tx1, 0, (size_t)(2 * BNC + Nn) * sizeof(float), stream);

    const int T = 256;
    k_deg      <<<(Ee + T - 1) / T, T, 0, stream>>>(ew, dst, deg);
    k_lapw     <<<(Ee + T - 1) / T, T, 0, stream>>>(ew, src, dst, deg, lapw);
    k_prep_cheb<<<((Cc / 2) * DIMH + T - 1) / T, T, 0, stream>>>(W_cheb, W0p_p, W1p_p, W2p_p);
    k_pack     <<<((DIMH / 2) * DIMH + T - 1) / T, T, 0, stream>>>(W1, W1_p, DIMH / 2, DIMH);
    k_pack     <<<((DIMH / 2) * Cc + T - 1) / T, T, 0, stream>>>(W2, W2_p, DIMH / 2, Cc);

    const long long propThreads = (long long)Ee * 64;
    const int propBlocks = (int)((propThreads + T - 1) / T);
    k_prop<<<propBlocks, T, 0, stream>>>(x,   Tx1, src, dst, lapw);  // Tx1 = L x
    k_prop<<<propBlocks, T, 0, stream>>>(Tx1, P2,  src, dst, lapw);  // P2  = L Tx1

    fused_mlp<<<BN / 16, 256, 0, stream>>>(x, Tx1, P2,
                                           W0p_p, W1p_p, W2p_p, b_cheb,
                                           W1_p, b1, W2_p, b2, beta,
                                           out_sum, out_fx);
}